// QuantumMeasurement_74474732913027
// MI455X (gfx1250) — compile-verified
//
#include <hip/hip_runtime.h>

// Quantum Pauli-Z expectation:  exp[b,q] = sum_i sign_q(i) * (re[b,i]^2 + im[b,i]^2)
// sign_q(i) = (-1)^{bit_(9-q)(i)}   (kron ordering: factor 0 is most significant).
// Computed as a [8192x1024] x [1024x16] fp32 GEMM on V_WMMA_F32_16X16X4_F32,
// with the +-1 "B" matrix synthesized in-register from the K index.
// Memory-bound: 64 MB of reads -> ~2.75 us floor @ 23.3 TB/s; WMMA fp32 keeps
// full precision at zero extra bandwidth cost.

typedef __attribute__((ext_vector_type(2))) float v2f;
typedef __attribute__((ext_vector_type(8))) float v8f;

namespace {
constexpr int kNQ    = 10;    // qubits (output columns)
constexpr int kD     = 1024;  // state dimension
constexpr int kWaves = 8;     // waves per block, each owns a K slice
constexpr int kBlock = kWaves * 32;
constexpr int kKSpan = kD / kWaves;  // 128 K per wave -> 32 WMMAs
constexpr int kTileM = 16;           // batch rows per block
}  // namespace

__global__ __launch_bounds__(kBlock) void qmeas_expz_wmma(
    const float* __restrict__ re,
    const float* __restrict__ im,
    float* __restrict__ out) {
  __shared__ float red[kWaves * kTileM * 16];  // 8 KB: per-wave 16x16 partials

  const int tid   = threadIdx.x;
  const int wave  = tid >> 5;
  const int lane  = tid & 31;
  const int mn    = lane & 15;          // A-matrix row (M) and B-matrix col (N)
  const int khalf = (lane >> 4) << 1;   // K sub-offset per half-wave: 0 or 2

  const int b0 = blockIdx.x * kTileM;
  const float* __restrict__ rowRe = re + (size_t)(b0 + mn) * kD;
  const float* __restrict__ rowIm = im + (size_t)(b0 + mn) * kD;

  const bool colValid = (mn < kNQ);
  const int  shift    = colValid ? (kNQ - 1 - mn) : 0;  // bit of k -> sign

  v8f c = {};  // fp32 16x16 accumulator (8 VGPRs, wave32 layout)
  const int kbase = wave * kKSpan + khalf;

#pragma unroll 4
  for (int kb = 0; kb < kKSpan; kb += 4) {
    const int k = kbase + kb;

    // A fragment: probabilities P[m][k], P[m][k+1] for this lane's row.
    const float2 r2 = *(const float2*)(rowRe + k);
    const float2 i2 = *(const float2*)(rowIm + k);
    v2f a;
    a.x = r2.x * r2.x + i2.x * i2.x;
    a.y = r2.y * r2.y + i2.y * i2.y;

    // B fragment: sign matrix S[k][n], S[k+1][n] for n = mn (0 for pad cols).
    const float sx = (((k)     >> shift) & 1) ? -1.0f : 1.0f;
    const float sy = (((k + 1) >> shift) & 1) ? -1.0f : 1.0f;
    v2f b;
    b.x = colValid ? sx : 0.0f;
    b.y = colValid ? sy : 0.0f;

    // D = A(16x4) x B(4x16) + C, fp32 end to end.
    c = __builtin_amdgcn_wmma_f32_16x16x4_f32(false, a, false, b, (short)0, c,
                                              false, false);
  }

  // Spill this wave's 16x16 partial tile to LDS.
  // C/D layout: VGPR r of lane l holds C[M = r + 8*(l>=16)][N = l&15].
#pragma unroll
  for (int r = 0; r < 8; ++r) {
    const int m = r + ((lane >> 4) << 3);
    red[wave * 256 + m * 16 + mn] = c[r];
  }
  __syncthreads();

  // Cross-wave K reduction: each of the 256 threads owns one (m, n) cell.
  float s = 0.0f;
#pragma unroll
  for (int w = 0; w < kWaves; ++w) s += red[w * 256 + tid];

  const int om = tid >> 4;
  const int on = tid & 15;
  if (on < kNQ) out[(size_t)(b0 + om) * kNQ + on] = s;
}

extern "C" void kernel_launch(void* const* d_in, const int* in_sizes, int n_in,
                              void* d_out, int out_size, void* d_ws, size_t ws_size,
                              hipStream_t stream) {
  (void)n_in; (void)out_size; (void)d_ws; (void)ws_size;
  const float* re = (const float*)d_in[0];   // state_real [8192, 1024] fp32
  const float* im = (const float*)d_in[1];   // state_imag [8192, 1024] fp32
  // d_in[2] (dense pauli_z_obs, 80 MB complex64) is deliberately unused:
  // Z_q is diagonal +-1 and is synthesized in-register.
  float* out = (float*)d_out;                // [8192, 10] fp32

  const int batch = in_sizes[0] / kD;        // 8192
  dim3 grid(batch / kTileM);                 // 512 blocks
  dim3 block(kBlock);                        // 256 threads = 8 waves
  hipLaunchKernelGGL(qmeas_expz_wmma, grid, block, 0, stream, re, im, out);
}